// MultiHeadAttention_57174604644739
// MI455X (gfx1250) — compile-verified
//
#include <hip/hip_runtime.h>
#include <hip/hip_bf16.h>
#include <math.h>

// ---------------------------------------------------------------------------
// Additive (Bahdanau) multi-head attention for MI455X / gfx1250.
//   bs=2, H=8, Lq=Lk=384, DQ=DK=DV=32, DC=64
// Stage 1: qW = Qh @ W1[:32,:], kWT = (Kh @ W1[32:,:])^T   (V_WMMA_F32_16X16X4_F32)
// Stage 2: scores = sum_e tanh(qW+kWT+b1)*w2, mask, softmax (hw v_tanh_f32,
//          8 q-rows per block for kW reuse)
// Stage 3: out = att @ Vh, query-length mask                (V_WMMA_F32_16X16X4_F32)
// ---------------------------------------------------------------------------

#define H_  8
#define DQ_ 32
#define DC_ 64
#define L_  384
#define HD_ 256   // H*32, row stride of q/k/v
#define QB_ 8     // q-rows per score block

typedef __attribute__((ext_vector_type(2))) float v2f;
typedef __attribute__((ext_vector_type(8))) float v8f;

__device__ __forceinline__ float tanh_hw(float x) {
#if __has_builtin(__builtin_amdgcn_tanhf)
    return __builtin_amdgcn_tanhf(x);
#elif __has_builtin(__builtin_amdgcn_tanh_f32)
    return __builtin_amdgcn_tanh_f32(x);
#else
    float y;
    // CDNA5 hardware transcendental; 2x v_nop covers the TRANS RAW hazard.
    asm volatile("v_tanh_f32 %0, %1\n\tv_nop\n\tv_nop" : "=v"(y) : "v"(x));
    return y;
#endif
}

// ---------------------------------------------------------------------------
// Kernel 1: projections via WMMA f32 16x16x4.
// grid = (96 = 24 mtiles * 4 etiles, 16 = bs*H, 2 = {q-proj, k-proj}), block = 32
// q-proj output: qW[(bh*384 + q)*64 + e]           (row major)
// k-proj output: kWT[(bh*64 + e)*384 + k]          (transposed for stage 2)
// ---------------------------------------------------------------------------
__global__ __launch_bounds__(32)
void proj_wmma_kernel(const float* __restrict__ q, const float* __restrict__ k,
                      const float* __restrict__ w1,
                      float* __restrict__ qW, float* __restrict__ kWT)
{
    const int lane  = threadIdx.x & 31;
    const int mtile = blockIdx.x >> 2;   // 0..23  (rows of Q/K, 16 each)
    const int etile = blockIdx.x & 3;    // 0..3   (cols of W1, 16 each)
    const int bh    = blockIdx.y;        // b*H + h
    const int h     = bh & 7;
    const int b     = bh >> 3;
    const int isK   = blockIdx.z;        // 0: q side, 1: k side

    const int n  = lane & 15;            // A-frag M index == B-frag N index
    const int hi = lane >> 4;            // K sub-pair select
    const int mb = mtile * 16;
    const int eb = etile * 16;

    const float* X    = isK ? k : q;                                      // (bs,384,256)
    const float* Xrow = X + ((size_t)b * L_ + (mb + n)) * HD_ + h * DQ_;  // A[m][*]
    const float* Wcol = w1 + ((size_t)h * DC_ + isK * DQ_) * DC_ + eb + n; // B[*][n]

    v8f c = {};
#pragma unroll
    for (int s = 0; s < 8; ++s) {        // K = 32 in steps of 4
        const int kb = s * 4 + 2 * hi;
        v2f a, bf;
        a.x  = Xrow[kb];
        a.y  = Xrow[kb + 1];
        bf.x = Wcol[(size_t)kb * DC_];
        bf.y = Wcol[(size_t)(kb + 1) * DC_];
        c = __builtin_amdgcn_wmma_f32_16x16x4_f32(false, a, false, bf,
                                                  (short)0, c, false, false);
    }

    // Branchless epilogue: uniform select of base pointer + per-r stride.
    //   q side: qW[(bh*384 + mb+8*hi + r)*64 + eb+n]   -> stride 64
    //   k side: kWT[(bh*64 + eb+n)*384 + mb+8*hi + r]  -> stride 1
    const size_t off_q = ((size_t)bh * L_ + mb + 8 * hi) * DC_ + (eb + n);
    const size_t off_k = ((size_t)bh * DC_ + eb + n) * L_ + (mb + 8 * hi);
    float* base = isK ? (kWT + off_k) : (qW + off_q);
    const int strideR = isK ? 1 : DC_;
#pragma unroll
    for (int r = 0; r < 8; ++r)
        base[(size_t)r * strideR] = c[r];
}

// ---------------------------------------------------------------------------
// Kernel 2: additive scores + masked softmax, 8 q-rows per block.
// grid = (48 = 384/8 q-tiles, 16 = bs*H), block = 384 (thread == key index k)
// Each thread loads kW[k,e] once and applies it to all 8 q-rows (LDS-cached
// qW+b1), giving 8 independent tanh/FMA chains (TRANS/VALU co-execution).
// ---------------------------------------------------------------------------
__global__ __launch_bounds__(384)
void score_softmax_kernel(const float* __restrict__ qW,
                          const float* __restrict__ kWT,
                          const float* __restrict__ b1,
                          const float* __restrict__ w2,
                          const int*   __restrict__ klen,
                          float* __restrict__ att)
{
    __shared__ float cqb[QB_][DC_];   // qW row + b1, for 8 q-rows
    __shared__ float sw2[DC_];
    __shared__ float red[512];

    const int tid = threadIdx.x;      // key index 0..383
    const int q0  = blockIdx.x * QB_;
    const int bh  = blockIdx.y;
    const int h   = bh & 7;
    const int b   = bh >> 3;

    for (int i = tid; i < QB_ * DC_; i += L_) {
        const int qi = i >> 6, e = i & 63;
        cqb[qi][e] = qW[((size_t)bh * L_ + q0 + qi) * DC_ + e] + b1[h * DC_ + e];
    }
    if (tid < DC_) sw2[tid] = w2[h * DC_ + tid];
    __syncthreads();

    const float* kcol = kWT + (size_t)bh * DC_ * L_ + tid;
    float acc[QB_] = {};
#pragma unroll 4
    for (int e = 0; e < DC_; ++e) {
        const float kv = kcol[(size_t)e * L_];
        const float w  = sw2[e];
#pragma unroll
        for (int qi = 0; qi < QB_; ++qi)
            acc[qi] += tanh_hw(cqb[qi][e] + kv) * w;
    }

    const bool valid = tid < klen[b];

    for (int qi = 0; qi < QB_; ++qi) {
        const float sc = valid ? acc[qi] : -1e9f;

        // block max (pad 384 -> 512 with -inf)
        red[tid] = sc;
        if (tid < 128) red[L_ + tid] = -3.4e38f;
        __syncthreads();
#pragma unroll
        for (int st = 256; st > 0; st >>= 1) {
            if (tid < st) red[tid] = fmaxf(red[tid], red[tid + st]);
            __syncthreads();
        }
        const float mx = red[0];
        __syncthreads();

        const float p = __expf(sc - mx);
        red[tid] = p;
        if (tid < 128) red[L_ + tid] = 0.f;
        __syncthreads();
#pragma unroll
        for (int st = 256; st > 0; st >>= 1) {
            if (tid < st) red[tid] += red[tid + st];
            __syncthreads();
        }
        const float inv = 1.f / red[0];

        att[((size_t)bh * L_ + q0 + qi) * L_ + tid] = p * inv;
        __syncthreads();   // red reused next row
    }
}

// ---------------------------------------------------------------------------
// Kernel 3: out = att @ V per (b,h), WMMA f32 16x16x4, K = 384.
// grid = (24 mtiles, 2 ntiles, 16 = bs*H), block = 32.
// out layout: (b, q, h, dv) -> b*384*256 + q*256 + h*32 + dv, qmask applied.
// ---------------------------------------------------------------------------
__global__ __launch_bounds__(32)
void av_wmma_kernel(const float* __restrict__ att, const float* __restrict__ v,
                    const int* __restrict__ qlen, float* __restrict__ out)
{
    const int lane  = threadIdx.x & 31;
    const int mtile = blockIdx.x;        // 0..23
    const int ntile = blockIdx.y;        // 0..1
    const int bh    = blockIdx.z;
    const int h     = bh & 7;
    const int b     = bh >> 3;

    const int n  = lane & 15;
    const int hi = lane >> 4;
    const int mb = mtile * 16;
    const int nb = ntile * 16;

    const float* Arow = att + ((size_t)bh * L_ + (mb + n)) * L_;          // A[m][*]
    const float* Bcol = v + (size_t)b * L_ * HD_ + h * DQ_ + nb + n;      // B[*][n]

    v8f c = {};
    for (int s = 0; s < 96; ++s) {       // K = 384 in steps of 4
        const int kb = s * 4 + 2 * hi;
        v2f a, bf;
        a.x  = Arow[kb];
        a.y  = Arow[kb + 1];
        bf.x = Bcol[(size_t)kb * HD_];
        bf.y = Bcol[(size_t)(kb + 1) * HD_];
        c = __builtin_amdgcn_wmma_f32_16x16x4_f32(false, a, false, bf,
                                                  (short)0, c, false, false);
    }

    const int ql = qlen[b];
#pragma unroll
    for (int r = 0; r < 8; ++r) {
        const int qrow  = mb + r + 8 * hi;
        const float val = (qrow < ql) ? c[r] : 0.f;
        out[((size_t)b * L_ + qrow) * HD_ + h * DQ_ + nb + n] = val;
    }
}

// ---------------------------------------------------------------------------
extern "C" void kernel_launch(void* const* d_in, const int* in_sizes, int n_in,
                              void* d_out, int out_size, void* d_ws, size_t ws_size,
                              hipStream_t stream) {
    const float* q   = (const float*)d_in[0];
    const float* k   = (const float*)d_in[1];
    const float* v   = (const float*)d_in[2];
    const int*   qsl = (const int*)d_in[3];
    const int*   ksl = (const int*)d_in[4];
    const float* w1  = (const float*)d_in[5];
    const float* b1  = (const float*)d_in[6];
    const float* w2  = (const float*)d_in[7];

    float* out = (float*)d_out;                        // (2,384,256) = 196608 floats
    float* att = (float*)d_out + 2 * L_ * HD_;         // (2,8,384,384)

    // workspace: qW (16*384*64) then kWT (16*64*384), 1.5 MB each
    float* qW  = (float*)d_ws;
    float* kWT = qW + (size_t)2 * H_ * L_ * DC_;

    proj_wmma_kernel<<<dim3(96, 16, 2), 32, 0, stream>>>(q, k, w1, qW, kWT);
    score_softmax_kernel<<<dim3(L_ / QB_, 16), L_, 0, stream>>>(qW, kWT, b1, w2, ksl, att);
    av_wmma_kernel<<<dim3(24, 2, 16), 32, 0, stream>>>(att, v, qsl, out);
}